// MambaMem_87136296501518
// MI455X (gfx1250) — compile-verified
//
#include <hip/hip_runtime.h>
#include <hip/hip_bf16.h>
#include <math.h>

// ---- problem constants (from reference) ----
#define B_   2
#define T_   12
#define N_   2048
#define D_   64
#define DI_  128
#define DS_  8
#define DC_  4
#define DTR_ 4
#define NM_  10
#define DM_  64
#define NK_  10
#define DK_  4
#define CK_  3
#define CS_  2
#define TC_  5      // (T-CK)/CS + 1
#define TT_  17     // T + TC
#define L_   (T_*N_)        // 24576
#define BL_  (B_*L_)        // 49152

typedef float v2f __attribute__((ext_vector_type(2)));
typedef float v8f __attribute__((ext_vector_type(8)));

// =====================================================================
// K0: fold small weight products:
//   WqMk[d][j]   = msam_Wq(64x4) @ msam_Mk.T(4x10)
//   memWqM[d][j] = mem_Wq(64x64) @ mem_M.T(64x10)
//   MWo[j][d]    = mem_M(10x64) @ mem_Wo(64x64)
// =====================================================================
__global__ void k_prep(const float* __restrict__ msam_Wq,
                       const float* __restrict__ msam_Mk,
                       const float* __restrict__ mem_Wq,
                       const float* __restrict__ mem_M,
                       const float* __restrict__ mem_Wo,
                       float* __restrict__ WqMk,
                       float* __restrict__ memWqM,
                       float* __restrict__ MWo) {
  int t = threadIdx.x;  // 64 threads
  if (t < D_) {
    for (int j = 0; j < NK_; ++j) {
      float s = 0.f;
      for (int k = 0; k < DK_; ++k) s += msam_Wq[t*DK_+k] * msam_Mk[j*DK_+k];
      WqMk[t*NK_+j] = s;
    }
    for (int j = 0; j < NM_; ++j) {
      float s = 0.f;
      for (int k = 0; k < DM_; ++k) s += mem_Wq[t*DM_+k] * mem_M[j*DM_+k];
      memWqM[t*NM_+j] = s;
    }
  }
  if (t < NM_) {
    for (int d = 0; d < D_; ++d) {
      float s = 0.f;
      for (int k = 0; k < DM_; ++k) s += mem_M[t*DM_+k] * mem_Wo[k*D_+d];
      MWo[t*D_+d] = s;
    }
  }
}

// =====================================================================
// K1: msam strided conv + attention.  One block per (b,n), 64 threads (d).
// conv[tc][d] = b[d] + sum_{k,din} x[b, tc*2+k, n, din] * w[k,din,d]
// att = softmax(conv @ WqMk);  conv_out = conv + att @ Mv
// =====================================================================
__global__ void __launch_bounds__(64) k_msam(const float* __restrict__ x,
                      const float* __restrict__ convw,
                      const float* __restrict__ convb,
                      const float* __restrict__ WqMk,
                      const float* __restrict__ Mv,
                      float* __restrict__ conv_out) {
  int bn = blockIdx.x;
  int b = bn / N_, n = bn % N_;
  int d = threadIdx.x;
  __shared__ float sx[11][D_];
  __shared__ float sc[TC_][D_];
  __shared__ float slog[TC_][NK_];
  __shared__ float satt[TC_][NK_];
  for (int t = 0; t < 11; ++t)
    sx[t][d] = x[(((size_t)b*T_ + t)*N_ + n)*D_ + d];
  __syncthreads();
  for (int tc = 0; tc < TC_; ++tc) {
    float acc = convb[d];
    for (int k = 0; k < CK_; ++k) {
      const float* xr = sx[tc*CS_ + k];
      for (int din = 0; din < D_; ++din)
        acc += xr[din] * convw[((size_t)k*D_ + din)*D_ + d];
    }
    sc[tc][d] = acc;
  }
  __syncthreads();
  int task = threadIdx.x;
  if (task < TC_*NK_) {
    int tc = task / NK_, j = task % NK_;
    float s = 0.f;
    for (int dd = 0; dd < D_; ++dd) s += sc[tc][dd] * WqMk[dd*NK_ + j];
    slog[tc][j] = s;
  }
  __syncthreads();
  if (task < TC_*NK_) {
    int tc = task / NK_, j = task % NK_;
    float m = slog[tc][0];
    for (int jj = 1; jj < NK_; ++jj) m = fmaxf(m, slog[tc][jj]);
    float den = 0.f;
    for (int jj = 0; jj < NK_; ++jj) den += __expf(slog[tc][jj] - m);
    satt[tc][j] = __expf(slog[tc][j] - m) / den;
  }
  __syncthreads();
  for (int tc = 0; tc < TC_; ++tc) {
    float acc = sc[tc][d];
    for (int j = 0; j < NK_; ++j) acc += satt[tc][j] * Mv[j*D_ + d];
    conv_out[(((size_t)b*TC_ + tc)*N_ + n)*D_ + d] = acc;
  }
}

// =====================================================================
// K2: fused memory attention + time mix + residual. Block per (b,n), 64 thr.
// For s in 0..16: row = xcat[b,s,n,:]; att2 = softmax(row @ memWqM);
//   memo = att2 @ MWo; acc[t] += Wt[t,s]*memo.  u = x + acc.
// =====================================================================
__global__ void __launch_bounds__(64) k_memo(const float* __restrict__ x,
                      const float* __restrict__ conv_out,
                      const float* __restrict__ memWqM,
                      const float* __restrict__ MWo,
                      const float* __restrict__ Wt,
                      float* __restrict__ u) {
  int bn = blockIdx.x;
  int b = bn / N_, n = bn % N_;
  int d = threadIdx.x;
  __shared__ float srow[D_];
  __shared__ float slog[NM_];
  __shared__ float satt[NM_];
  float acc[T_];
  for (int t = 0; t < T_; ++t) acc[t] = 0.f;
  for (int s = 0; s < TT_; ++s) {
    float rv = (s < T_) ? x[(((size_t)b*T_ + s)*N_ + n)*D_ + d]
                        : conv_out[(((size_t)b*TC_ + (s - T_))*N_ + n)*D_ + d];
    srow[d] = rv;
    __syncthreads();
    if (d < NM_) {
      float sum = 0.f;
      for (int dd = 0; dd < D_; ++dd) sum += srow[dd] * memWqM[dd*NM_ + d];
      slog[d] = sum;
    }
    __syncthreads();
    if (d < NM_) {
      float m = slog[0];
      for (int j = 1; j < NM_; ++j) m = fmaxf(m, slog[j]);
      float den = 0.f;
      for (int j = 0; j < NM_; ++j) den += __expf(slog[j] - m);
      satt[d] = __expf(slog[d] - m) / den;
    }
    __syncthreads();
    float memo_d = 0.f;
    for (int j = 0; j < NM_; ++j) memo_d += satt[j] * MWo[j*D_ + d];
    for (int t = 0; t < T_; ++t) acc[t] += Wt[t*TT_ + s] * memo_d;
    __syncthreads();
  }
  for (int t = 0; t < T_; ++t) {
    size_t idx = (((size_t)b*T_ + t)*N_ + n)*D_ + d;
    u[idx] = x[idx] + acc[t];
  }
}

// =====================================================================
// K3: LayerNorm over D=64, one wave32 per row (2 elems per lane).
// =====================================================================
__global__ void __launch_bounds__(256) k_ln(const float* __restrict__ u,
                    const float* __restrict__ lnw,
                    const float* __restrict__ lnb,
                    float* __restrict__ un) {
  int wave = threadIdx.x >> 5, lane = threadIdx.x & 31;
  size_t row = (size_t)blockIdx.x * 8 + wave;
  const float* ur = u + row * D_;
  float a0 = ur[lane], a1 = ur[lane + 32];
  float s = a0 + a1, sq = a0*a0 + a1*a1;
  for (int off = 16; off > 0; off >>= 1) {
    s  += __shfl_xor(s,  off, 32);
    sq += __shfl_xor(sq, off, 32);
  }
  float mean = s * (1.f/64.f);
  float var  = sq * (1.f/64.f) - mean*mean;
  float inv  = rsqrtf(var + 1e-5f);
  float* o = un + row * D_;
  o[lane]      = (a0 - mean) * inv * lnw[lane]      + lnb[lane];
  o[lane + 32] = (a1 - mean) * inv * lnw[lane + 32] + lnb[lane + 32];
}

// =====================================================================
// fp32 WMMA 16x16x4 fragment loaders (wave32).
// A (16x4): lanes 0-15 rows, VGPR0/1 = K pair {k,k+1}; lanes 16-31 = {k+2,k+3}
// B (4x16): lanes hold N cols; VGPR0/1 = K rows mirrored.
// C/D (16x16 f32, 8 VGPRs): VGPR g -> row g (+8 for lanes 16-31), col = lane&15
// =====================================================================
__device__ __forceinline__ v2f load_a_frag(const float* __restrict__ A, int lda,
                                           int row0, int k, int lane) {
  int r  = row0 + (lane & 15);
  int kk = k + ((lane >> 4) << 1);
  const float* p = A + (size_t)r * lda + kk;
  v2f a; a.x = p[0]; a.y = p[1];
  return a;
}
__device__ __forceinline__ v2f load_b_frag(const float* __restrict__ Bw, int ldb,
                                           int k, int col0, int lane) {
  int c  = col0 + (lane & 15);
  int kk = k + ((lane >> 4) << 1);
  v2f bv;
  bv.x = Bw[(size_t)kk       * ldb + c];
  bv.y = Bw[(size_t)(kk + 1) * ldb + c];
  return bv;
}

// =====================================================================
// K4: xz = un(BLx64) @ in_proj_w(64x256); cols<128 -> xmraw, else -> z
// WMMA fp32, grid (BL/16, 4), block 128 (4 waves, one 16-col tile each)
// =====================================================================
__global__ void __launch_bounds__(128) k_inproj(const float* __restrict__ un,
                                                const float* __restrict__ W,
                                                float* __restrict__ xmraw,
                                                float* __restrict__ zb) {
  int lane = threadIdx.x & 31;
  int wv   = threadIdx.x >> 5;
  int row0 = blockIdx.x * 16;
  int tile_n = blockIdx.y * 4 + wv;   // 0..15
  int col0 = tile_n * 16;
  v8f c = {};
#pragma unroll
  for (int k = 0; k < D_; k += 4) {
    v2f a  = load_a_frag(un, D_, row0, k, lane);
    v2f bf = load_b_frag(W, 2*DI_, k, col0, lane);
    c = __builtin_amdgcn_wmma_f32_16x16x4_f32(false, a, false, bf,
                                              (short)0, c, false, false);
  }
  float* dst = (tile_n < 8) ? xmraw : zb;
  int cbase  = (tile_n < 8) ? col0 : (col0 - DI_);
  int col    = cbase + (lane & 15);
#pragma unroll
  for (int g = 0; g < 8; ++g) {
    int r = row0 + g + ((lane >> 4) << 3);
    dst[(size_t)r * DI_ + col] = c[g];
  }
}

// =====================================================================
// K5: depthwise causal conv (K=4, pad 3 left) + bias + SiLU
// =====================================================================
__global__ void k_dwconv(const float* __restrict__ xmraw,
                         const float* __restrict__ w,
                         const float* __restrict__ bias,
                         float* __restrict__ xm) {
  size_t i = (size_t)blockIdx.x * blockDim.x + threadIdx.x;
  if (i >= (size_t)BL_ * DI_) return;
  int di = (int)(i % DI_);
  size_t bl = i / DI_;
  int l = (int)(bl % L_);
  float acc = bias[di];
#pragma unroll
  for (int k = 0; k < DC_; ++k) {
    int ls = l - (DC_ - 1) + k;
    if (ls >= 0)
      acc += w[k*DI_ + di] * xmraw[(bl + (size_t)(k - (DC_-1)))*DI_ + di];
  }
  xm[i] = acc / (1.f + __expf(-acc));
}

// =====================================================================
// K6: proj = xm(BLx128) @ x_proj_w(128x20)
// =====================================================================
__global__ void k_xproj(const float* __restrict__ xm,
                        const float* __restrict__ W,
                        float* __restrict__ proj) {
  size_t i = (size_t)blockIdx.x * blockDim.x + threadIdx.x;
  if (i >= (size_t)BL_ * (DTR_ + 2*DS_)) return;
  int j = (int)(i % (DTR_ + 2*DS_));
  size_t row = i / (DTR_ + 2*DS_);
  const float* xr = xm + row * DI_;
  float s = 0.f;
  for (int k = 0; k < DI_; ++k) s += xr[k] * W[k*(DTR_ + 2*DS_) + j];
  proj[i] = s;
}

// =====================================================================
// K7: selective scan.  Block per batch, 128 threads (one per di).
// dt = softplus(proj[:,:4] @ dt_proj_w + b); h = exp(dt*A)*h + dt*xm*Bm;
// y = sum_s h*Cm + xm*Dp; y *= silu(z)
// =====================================================================
__global__ void __launch_bounds__(DI_) k_scan(const float* __restrict__ proj,
                     const float* __restrict__ xm,
                     const float* __restrict__ zbuf,
                     const float* __restrict__ dtw,
                     const float* __restrict__ dtb,
                     const float* __restrict__ A_log,
                     const float* __restrict__ Dp,
                     float* __restrict__ y) {
  int b  = blockIdx.x;
  int di = threadIdx.x;
  __shared__ float sp[DTR_ + 2*DS_];
  float Aa[DS_];
#pragma unroll
  for (int s = 0; s < DS_; ++s) Aa[s] = -__expf(A_log[di*DS_ + s]);
  float w0 = dtw[0*DI_+di], w1 = dtw[1*DI_+di], w2 = dtw[2*DI_+di], w3 = dtw[3*DI_+di];
  float bb = dtb[di];
  float dpv = Dp[di];
  float h[DS_];
#pragma unroll
  for (int s = 0; s < DS_; ++s) h[s] = 0.f;
  for (int l = 0; l < L_; ++l) {
    size_t row = (size_t)b * L_ + l;
    if (di < DTR_ + 2*DS_) sp[di] = proj[row*(DTR_ + 2*DS_) + di];
    __syncthreads();
    float dtin = sp[0]*w0 + sp[1]*w1 + sp[2]*w2 + sp[3]*w3 + bb;
    float dt = (dtin > 20.f) ? dtin : log1pf(__expf(dtin));
    float xv = xm[row*DI_ + di];
    float dtx = dt * xv;
    float yv = 0.f;
#pragma unroll
    for (int s = 0; s < DS_; ++s) {
      float dA = __expf(dt * Aa[s]);
      h[s] = dA * h[s] + dtx * sp[DTR_ + s];
      yv  += h[s] * sp[DTR_ + DS_ + s];
    }
    yv += xv * dpv;
    float zv = zbuf[row*DI_ + di];
    yv *= zv / (1.f + __expf(-zv));
    y[row*DI_ + di] = yv;
    __syncthreads();
  }
}

// =====================================================================
// K8: out = u + y(BLx128) @ out_proj_w(128x64).  WMMA fp32.
// grid (BL/16), block 128 (4 waves = 4 col tiles of 16)
// =====================================================================
__global__ void __launch_bounds__(128) k_outproj(const float* __restrict__ y,
                                                 const float* __restrict__ W,
                                                 const float* __restrict__ u,
                                                 float* __restrict__ out) {
  int lane = threadIdx.x & 31;
  int wv   = threadIdx.x >> 5;
  int row0 = blockIdx.x * 16;
  int col0 = wv * 16;
  v8f c = {};
#pragma unroll
  for (int k = 0; k < DI_; k += 4) {
    if (k + 16 < DI_)
      __builtin_prefetch(y + (size_t)(row0 + (lane & 15)) * DI_ + k + 16, 0, 1);
    v2f a  = load_a_frag(y, DI_, row0, k, lane);
    v2f bf = load_b_frag(W, D_, k, col0, lane);
    c = __builtin_amdgcn_wmma_f32_16x16x4_f32(false, a, false, bf,
                                              (short)0, c, false, false);
  }
  int col = col0 + (lane & 15);
#pragma unroll
  for (int g = 0; g < 8; ++g) {
    int r = row0 + g + ((lane >> 4) << 3);
    size_t idx = (size_t)r * D_ + col;
    out[idx] = u[idx] + c[g];
  }
}

// =====================================================================
extern "C" void kernel_launch(void* const* d_in, const int* in_sizes, int n_in,
                              void* d_out, int out_size, void* d_ws, size_t ws_size,
                              hipStream_t stream) {
  const float* x           = (const float*)d_in[0];
  const float* msam_conv_w = (const float*)d_in[1];
  const float* msam_conv_b = (const float*)d_in[2];
  const float* msam_Wq     = (const float*)d_in[3];
  const float* msam_Mk     = (const float*)d_in[4];
  const float* msam_Mv     = (const float*)d_in[5];
  const float* mem_Wq      = (const float*)d_in[6];
  const float* mem_M       = (const float*)d_in[7];
  const float* mem_Wo      = (const float*)d_in[8];
  const float* mem_Wt      = (const float*)d_in[9];
  const float* ln_w        = (const float*)d_in[10];
  const float* ln_b        = (const float*)d_in[11];
  const float* in_proj_w   = (const float*)d_in[12];
  const float* conv1d_w    = (const float*)d_in[13];
  const float* conv1d_b    = (const float*)d_in[14];
  const float* x_proj_w    = (const float*)d_in[15];
  const float* dt_proj_w   = (const float*)d_in[16];
  const float* dt_proj_b   = (const float*)d_in[17];
  const float* A_log       = (const float*)d_in[18];
  const float* Dp          = (const float*)d_in[19];
  const float* out_proj_w  = (const float*)d_in[20];
  float* out = (float*)d_out;

  // workspace layout (floats)
  float* ws = (float*)d_ws;
  float* WqMk     = ws;                 // 64*10
  float* memWqM   = WqMk + 640;         // 64*10
  float* MWo      = memWqM + 640;       // 10*64
  float* conv_out = ws + 2048;                               // B*TC*N*D = 1,310,720
  float* u        = conv_out + (size_t)B_*TC_*N_*D_;         // BL*64
  float* un       = u        + (size_t)BL_*D_;               // BL*64  (aliased by proj later)
  float* xmraw    = un       + (size_t)BL_*D_;               // BL*128 (aliased by y later)
  float* zb       = xmraw    + (size_t)BL_*DI_;              // BL*128
  float* xm       = zb       + (size_t)BL_*DI_;              // BL*128
  float* proj     = un;      // un dead after k_inproj
  float* ybuf     = xmraw;   // xmraw dead after k_dwconv

  k_prep<<<1, 64, 0, stream>>>(msam_Wq, msam_Mk, mem_Wq, mem_M, mem_Wo,
                               WqMk, memWqM, MWo);
  k_msam<<<B_*N_, 64, 0, stream>>>(x, msam_conv_w, msam_conv_b, WqMk, msam_Mv,
                                   conv_out);
  k_memo<<<B_*N_, 64, 0, stream>>>(x, conv_out, memWqM, MWo, mem_Wt, u);
  k_ln<<<BL_/8, 256, 0, stream>>>(u, ln_w, ln_b, un);
  k_inproj<<<dim3(BL_/16, 4), 128, 0, stream>>>(un, in_proj_w, xmraw, zb);
  k_dwconv<<<(BL_*DI_)/256, 256, 0, stream>>>(xmraw, conv1d_w, conv1d_b, xm);
  k_xproj<<<((size_t)BL_*(DTR_+2*DS_) + 255)/256, 256, 0, stream>>>(xm, x_proj_w, proj);
  k_scan<<<B_, DI_, 0, stream>>>(proj, xm, zb, dt_proj_w, dt_proj_b, A_log, Dp, ybuf);
  k_outproj<<<BL_/16, 128, 0, stream>>>(ybuf, out_proj_w, u, out);
}